// InteractionMapBuilder_12386685682254
// MI455X (gfx1250) — compile-verified
//
#include <hip/hip_runtime.h>
#include <math.h>

typedef __attribute__((ext_vector_type(2))) float v2f;
typedef __attribute__((ext_vector_type(4))) float v4f;
typedef __attribute__((ext_vector_type(8))) float v8f;

#define B_    2
#define L_    512
#define DH    1024
#define P_    64
#define MTOT  (B_ * L_)        // 1024 rows per projected input
#define NCH   (4 * P_ + 1)     // 257 output channels
#define EPS_  1e-8f

__device__ __forceinline__ float gelu_exact(float x) {
    // torch default (exact) GELU: 0.5*x*(1+erf(x/sqrt(2)))
    return 0.5f * x * (1.0f + erff(x * 0.70710678118654752440f));
}

// ---------------------------------------------------------------------------
// Kernel 1: z = gelu(h @ W + bias).  h: [MTOT, DH] (B,L flattened), W: [DH, P_],
// z: [MTOT, P_].  One wave per 16x16 output tile; K swept 4 at a time with
// V_WMMA_F32_16X16X4_F32.
//
// Fragment layouts (CDNA5 ISA 7.12.2, 32-bit):
//   A 16x4:  lane l (l<16): v0=A[m=l][k+0], v1=A[m=l][k+1]
//            lane l (>=16): v0=A[m=l-16][k+2], v1=A[m=l-16][k+3]
//   B 4x16:  lane l (l<16): v0=B[k+0][n=l],   v1=B[k+1][n=l]
//            lane l (>=16): v0=B[k+2][n=l-16],v1=B[k+3][n=l-16]
//   C/D 16x16: vgpr r: lanes0-15 -> M=r, N=lane; lanes16-31 -> M=r+8, N=lane-16
// ---------------------------------------------------------------------------
__global__ __launch_bounds__(32) void proj_gelu_wmma(
    const float* __restrict__ h, const float* __restrict__ W,
    const float* __restrict__ bias, float* __restrict__ z)
{
    const int tile = blockIdx.x;          // 64 m-tiles x 4 n-tiles = 256
    const int m0   = (tile >> 2) << 4;
    const int n0   = (tile & 3) << 4;
    const int lane = threadIdx.x;
    const int l16  = lane & 15;
    const int half = lane >> 4;

    const float* ap = h + (size_t)(m0 + l16) * DH + half * 2;     // A row base
    const float* bp = W + (size_t)(half * 2) * P_ + n0 + l16;     // B col base

    v8f acc = {};
    for (int k = 0; k < DH; k += 4) {
        v2f a = *(const v2f*)(ap + k);            // A[m][k+2h], A[m][k+2h+1]
        v2f b;
        b.x = bp[(size_t)k * P_];                 // B[k+2h  ][n]
        b.y = bp[(size_t)k * P_ + P_];            // B[k+2h+1][n]
        acc = __builtin_amdgcn_wmma_f32_16x16x4_f32(
                  false, a, false, b, (short)0, acc, false, false);
    }

    const float bc = bias[n0 + l16];
#pragma unroll
    for (int r = 0; r < 8; ++r) {
        const int row = m0 + r + half * 8;
        float g = gelu_exact(acc[r] + bc);
        z[(size_t)row * P_ + n0 + l16] = g;
    }
}

// ---------------------------------------------------------------------------
// Kernel 2: per-row sum of squares (deterministic, no atomics).
// z_a and z_b are contiguous in the workspace -> one launch covers both.
// ---------------------------------------------------------------------------
__global__ __launch_bounds__(256) void rownorm_kernel(
    const float* __restrict__ z, float* __restrict__ sumsq, int nrows)
{
    int row = blockIdx.x * blockDim.x + threadIdx.x;
    if (row >= nrows) return;
    const float* p = z + (size_t)row * P_;
    float s = 0.f;
#pragma unroll
    for (int c = 0; c < P_; c += 4) {
        v4f v = *(const v4f*)(p + c);
        s += v.x * v.x + v.y * v.y + v.z * v.z + v.w * v.w;
    }
    sumsq[row] = s;
}

// ---------------------------------------------------------------------------
// Kernel 3: interaction map.  Block = 128 threads (4 waves), tile = 16 i x 64 j.
// Phase A: each wave computes a 16x16 dot tile (K=64) with wmma from LDS and
//          writes the cosine-sim channel (c=256).
// Phase B: 256 elementwise channels streamed as float4 non-temporal stores.
// ---------------------------------------------------------------------------
#define PAD 68   // row pitch in floats: 272B -> 16B aligned rows, bank-spread

__global__ __launch_bounds__(128) void interact_kernel(
    const float* __restrict__ z_a, const float* __restrict__ z_b,
    const float* __restrict__ sa,  const float* __restrict__ sb,
    float* __restrict__ out)
{
    __shared__ float za [16][PAD];   // z_a tile, [i][c]
    __shared__ float zb [64][PAD];   // z_b tile, [j][c]   (wmma B fragments)
    __shared__ float zbt[64][PAD];   // z_b tile, [c][j]   (elementwise phase)

    const int j0 = blockIdx.x * 64;
    const int i0 = blockIdx.y * 16;
    const int bb = blockIdx.z;
    const int tid = threadIdx.x;

    const float* Za = z_a + ((size_t)bb * L_) * P_;
    const float* Zb = z_b + ((size_t)bb * L_) * P_;

    // Cooperative loads
    for (int idx = tid; idx < 16 * 64; idx += 128) {
        int i = idx >> 6, c = idx & 63;
        za[i][c] = Za[(size_t)(i0 + i) * P_ + c];
    }
    for (int idx = tid; idx < 64 * 64; idx += 128) {
        int j = idx >> 6, c = idx & 63;
        float v = Zb[(size_t)(j0 + j) * P_ + c];
        zb[j][c]  = v;
        zbt[c][j] = v;
    }
    __syncthreads();

    const size_t chstride = (size_t)L_ * L_;
    const size_t obase    = (size_t)bb * NCH * chstride;

    // ---- Phase A: wmma dot -> sim channel -------------------------------
    {
        const int wave = tid >> 5;            // 0..3 -> j sub-tile
        const int lane = tid & 31;
        const int l16  = lane & 15;
        const int half = lane >> 4;

        v8f acc = {};
#pragma unroll
        for (int k = 0; k < P_; k += 4) {
            v2f a = *(const v2f*)&za[l16][k + half * 2];
            v2f b = *(const v2f*)&zb[wave * 16 + l16][k + half * 2];
            acc = __builtin_amdgcn_wmma_f32_16x16x4_f32(
                      false, a, false, b, (short)0, acc, false, false);
        }
        const int j = j0 + wave * 16 + l16;
        const float rb = sb[bb * L_ + j] + EPS_;
#pragma unroll
        for (int r = 0; r < 8; ++r) {
            const int i = i0 + r + half * 8;
            float sim = acc[r] * rsqrtf((sa[bb * L_ + i] + EPS_) * rb);
            __builtin_nontemporal_store(
                sim, out + obase + (size_t)256 * chstride + (size_t)i * L_ + j);
        }
    }

    // ---- Phase B: elementwise channels (za_e, zb_e, diff, prod) ---------
    {
        const int ii   = tid >> 3;            // 0..15 local i
        const int jloc = (tid & 7) * 8;       // local j, 8 per thread
        const size_t rowoff = (size_t)(i0 + ii) * L_ + (j0 + jloc);
        float* o0 = out + obase + rowoff;                       // za_e  base
        float* o1 = o0 + (size_t)P_  * chstride;                // zb_e
        float* o2 = o0 + (size_t)(2 * P_) * chstride;           // diff
        float* o3 = o0 + (size_t)(3 * P_) * chstride;           // prod

        for (int c = 0; c < P_; ++c) {
            const float a  = za[ii][c];
            const v4f  b0  = *(const v4f*)&zbt[c][jloc];
            const v4f  b1  = *(const v4f*)&zbt[c][jloc + 4];
            const v4f  av  = {a, a, a, a};
            const size_t co = (size_t)c * chstride;

            __builtin_nontemporal_store(av, (v4f*)(o0 + co));
            __builtin_nontemporal_store(av, (v4f*)(o0 + co + 4));
            __builtin_nontemporal_store(b0, (v4f*)(o1 + co));
            __builtin_nontemporal_store(b1, (v4f*)(o1 + co + 4));

            v4f d0 = {fabsf(a - b0.x), fabsf(a - b0.y), fabsf(a - b0.z), fabsf(a - b0.w)};
            v4f d1 = {fabsf(a - b1.x), fabsf(a - b1.y), fabsf(a - b1.z), fabsf(a - b1.w)};
            __builtin_nontemporal_store(d0, (v4f*)(o2 + co));
            __builtin_nontemporal_store(d1, (v4f*)(o2 + co + 4));

            v4f p0 = {a * b0.x, a * b0.y, a * b0.z, a * b0.w};
            v4f p1 = {a * b1.x, a * b1.y, a * b1.z, a * b1.w};
            __builtin_nontemporal_store(p0, (v4f*)(o3 + co));
            __builtin_nontemporal_store(p1, (v4f*)(o3 + co + 4));
        }
    }
}

// ---------------------------------------------------------------------------
extern "C" void kernel_launch(void* const* d_in, const int* in_sizes, int n_in,
                              void* d_out, int out_size, void* d_ws, size_t ws_size,
                              hipStream_t stream)
{
    const float* h_a = (const float*)d_in[0];   // [2,512,1024]
    const float* h_b = (const float*)d_in[1];   // [2,512,1024]
    const float* Wa  = (const float*)d_in[2];   // [1024,64]
    const float* ba  = (const float*)d_in[3];   // [64]
    const float* Wb  = (const float*)d_in[4];   // [1024,64]
    const float* bb  = (const float*)d_in[5];   // [64]
    float* out = (float*)d_out;                 // [2,257,512,512]

    // Workspace layout (floats): z_a | z_b | sa | sb
    float* z_a = (float*)d_ws;
    float* z_b = z_a + (size_t)MTOT * P_;
    float* sa  = z_b + (size_t)MTOT * P_;
    float* sb  = sa + MTOT;

    // 1) projections (one wave per 16x16 tile: 64 m-tiles * 4 n-tiles = 256)
    proj_gelu_wmma<<<256, 32, 0, stream>>>(h_a, Wa, ba, z_a);
    proj_gelu_wmma<<<256, 32, 0, stream>>>(h_b, Wb, bb, z_b);

    // 2) row norms for both z_a and z_b (contiguous -> single launch)
    rownorm_kernel<<<(2 * MTOT + 255) / 256, 256, 0, stream>>>(z_a, sa, 2 * MTOT);
    (void)sb; // sb == sa + MTOT, covered by the launch above

    // 3) interaction map: 8 j-tiles x 32 i-tiles x 2 batches
    dim3 grid(L_ / 64, L_ / 16, B_);
    interact_kernel<<<grid, 128, 0, stream>>>(z_a, z_b, sa, sb, out);
}